// CRF_89258010346242
// MI455X (gfx1250) — compile-verified
//
#include <hip/hip_runtime.h>
#include <math.h>

typedef __attribute__((ext_vector_type(2))) float v2f;
typedef __attribute__((ext_vector_type(8))) float v8f;

#define BATCH   128
#define SEQ     512
#define TAGS    50      // TAG_SIZE
#define TP      64      // padded tag dim (multiple of 16)
#define KTILES  13      // ceil(TAGS/4): K rows 52..63 are identically zero
#define START   48      // TAG_SIZE-2
#define STOP    49      // TAG_SIZE-1
#define NEGINF  (-__builtin_inff())

// ---------------------------------------------------------------------------
// Forward (log-partition) scan.  One workgroup = 16 batch rows, 4 waves; each
// wave owns a 16-wide tag-column tile.  B operand (exp(transitions), zero
// padded) lives in 13 x v2f = 26 VGPRs per lane for all 511 steps.  Recurrent
// state P and scratch U live in LDS.  Row-max is a 3-step __shfl_xor within
// each wave (rows are contiguous 8-lane groups), so the scan needs only two
// workgroup barriers per step.
// ---------------------------------------------------------------------------
__global__ __launch_bounds__(128)
void crf_fwd_89258010346242(const float* __restrict__ em,
                            const int*   __restrict__ mask,
                            const float* __restrict__ tr,
                            float*       __restrict__ out)
{
    __shared__ float P[16 * TP];     // log partition, padded cols = -inf
    __shared__ float U[16 * TP];     // exp(P - rowmax), padded cols = 0
    __shared__ float cmax[16];
    __shared__ int   rmask[16];

    const int tid   = threadIdx.x;
    const int lane  = tid & 31;
    const int wv    = tid >> 5;          // wave id == N tile id (0..3)
    const int b0    = blockIdx.x * 16;   // first batch row of this tile
    const int r     = tid >> 3;          // row (0..15) for phase-1 work
    const int sub   = tid & 7;           // 8 threads cooperate per row

    const int jcol  = 16 * wv + (lane & 15);  // this lane's B/C/D column
    const int khalf = 2 * (lane >> 4);        // K sub-pair per lane half
    const int mrow  = lane & 15;              // this lane's A row

    // ---- init: P[r][j] = em[b,0,j] + trans[START][j]; pads: P=-inf, U=0 ---
    for (int k = 0; k < 8; ++k) {
        int j = sub + 8 * k;
        if (j < TAGS) {
            P[r * TP + j] = em[(size_t)(b0 + r) * (SEQ * TAGS) + j]
                            + tr[START * TAGS + j];
        } else {
            P[r * TP + j] = NEGINF;
            U[r * TP + j] = 0.f;         // written once, never touched again
        }
    }

    // ---- preload B fragments: E = exp(trans), zero-padded ------------------
    v2f Bf[KTILES];
#pragma unroll
    for (int f = 0; f < KTILES; ++f) {
        int k0 = 4 * f + khalf;
        float e0 = 0.f, e1 = 0.f;
        if (jcol < TAGS) {
            if (k0 < TAGS)     e0 = expf(tr[k0 * TAGS + jcol]);
            if (k0 + 1 < TAGS) e1 = expf(tr[(k0 + 1) * TAGS + jcol]);
        }
        Bf[f].x = e0;
        Bf[f].y = e1;
    }

    __syncthreads();

    for (int t = 1; t < SEQ; ++t) {
        // -- phase 1: row max via intra-wave shuffles, then U = exp(P - c) ---
        float m = NEGINF;
#pragma unroll
        for (int k = 0; k < 8; ++k) {
            int j = sub + 8 * k;
            if (j < TAGS) m = fmaxf(m, P[r * TP + j]);
        }
        m = fmaxf(m, __shfl_xor(m, 1, 8));
        m = fmaxf(m, __shfl_xor(m, 2, 8));
        m = fmaxf(m, __shfl_xor(m, 4, 8));
        if (sub == 0) {
            cmax[r]  = m;
            rmask[r] = mask[(b0 + r) * SEQ + t];
        }
#pragma unroll
        for (int k = 0; k < 8; ++k) {
            int j = sub + 8 * k;
            if (j < TAGS) U[r * TP + j] = expf(P[r * TP + j] - m);
        }
        __syncthreads();                 // U, cmax, rmask ready

        // -- phase 2: V = U x E via 13 chained f32 16x16x4 WMMAs -------------
        v8f acc = {0.f, 0.f, 0.f, 0.f, 0.f, 0.f, 0.f, 0.f};
#pragma unroll
        for (int f = 0; f < KTILES; ++f) {
            v2f a = *(const v2f*)&U[mrow * TP + 4 * f + khalf];
            acc = __builtin_amdgcn_wmma_f32_16x16x4_f32(
                /*neg_a=*/false, a, /*neg_b=*/false, Bf[f],
                /*c_mod=*/(short)0, acc, /*reuse_a=*/false, /*reuse_b=*/false);
        }

        // -- phase 3: P[b,j] = mask ? e_t + c + log(V) : P  ------------------
#pragma unroll
        for (int v = 0; v < 8; ++v) {
            int rr = v + 8 * (lane >> 4);       // C/D row for this VGPR/lane
            if (jcol < TAGS) {
                float pv = P[rr * TP + jcol];
                float nv = pv;
                if (rmask[rr]) {
                    nv = em[(size_t)(b0 + rr) * (SEQ * TAGS)
                            + (size_t)t * TAGS + jcol]
                         + cmax[rr] + logf(acc[v]);
                }
                P[rr * TP + jcol] = nv;
            }
        }
        __syncthreads();                 // P stable for next step's phase 1
    }

    // ---- final: sum_b logsumexp_i( P[b,i] + trans[i,STOP] ) ----------------
    if (tid < 16) {
        float mx = NEGINF;
        for (int i = 0; i < TAGS; ++i)
            mx = fmaxf(mx, P[tid * TP + i] + tr[i * TAGS + STOP]);
        float s = 0.f;
        for (int i = 0; i < TAGS; ++i)
            s += expf(P[tid * TP + i] + tr[i * TAGS + STOP] - mx);
        atomicAdd(out, mx + logf(s));
    }
}

// ---------------------------------------------------------------------------
// Gold-path score (subtracted from the output): pure gather + reduction.
// ---------------------------------------------------------------------------
__global__ __launch_bounds__(128)
void crf_gold_89258010346242(const float* __restrict__ em,
                             const int*   __restrict__ mask,
                             const int*   __restrict__ tags,
                             const float* __restrict__ tr,
                             float*       __restrict__ out)
{
    __shared__ float gs[128];
    __shared__ int   cs[128];
    const int b = blockIdx.x, tid = threadIdx.x;

    float g = 0.f;
    int cnt = 0;
    for (int s = tid; s < SEQ; s += 128) {
        if (mask[b * SEQ + s]) {
            int tg = tags[b * SEQ + s];
            int pv = (s == 0) ? START : tags[b * SEQ + s - 1];
            g += em[(size_t)b * SEQ * TAGS + (size_t)s * TAGS + tg]
                 + tr[pv * TAGS + tg];
            cnt++;
        }
    }
    gs[tid] = g;
    cs[tid] = cnt;
    __syncthreads();
    for (int off = 64; off > 0; off >>= 1) {
        if (tid < off) { gs[tid] += gs[tid + off]; cs[tid] += cs[tid + off]; }
        __syncthreads();
    }
    if (tid == 0) {
        int len  = cs[0];                         // mask is contiguous prefix
        int endt = tags[b * SEQ + len - 1];
        atomicAdd(out, -(gs[0] + tr[endt * TAGS + STOP]));
    }
}

__global__ void crf_zero_89258010346242(float* out) { out[0] = 0.f; }

// ---------------------------------------------------------------------------
extern "C" void kernel_launch(void* const* d_in, const int* in_sizes, int n_in,
                              void* d_out, int out_size, void* d_ws, size_t ws_size,
                              hipStream_t stream)
{
    const float* em   = (const float*)d_in[0];   // [128,512,50] f32
    const int*   mask = (const int*)  d_in[1];   // [128,512]
    const int*   tags = (const int*)  d_in[2];   // [128,512]
    const float* tr   = (const float*)d_in[3];   // [50,50] f32
    float*       out  = (float*)d_out;           // scalar

    crf_zero_89258010346242<<<1, 1, 0, stream>>>(out);
    crf_gold_89258010346242<<<BATCH, 128, 0, stream>>>(em, mask, tags, tr, out);
    crf_fwd_89258010346242<<<BATCH / 16, 128, 0, stream>>>(em, mask, tr, out);
}